// MultiBoxLoss_14370960572532
// MI455X (gfx1250) — compile-verified
//
#include <hip/hip_runtime.h>
#include <math.h>

// ---------------------------------------------------------------------------
// MultiBoxLoss for MI455X (gfx1250).
// Roofline: dominant cost is streaming conf_data (32*32768*81 f32 = ~340MB)
// once -> ~15us at 23.3 TB/s. Everything else (67M IoUs, radix-select) is
// cheap VALU work sized to hide under that stream. conf tiles are staged to
// LDS with GLOBAL_LOAD_ASYNC_TO_LDS_B128 (ASYNCcnt), block reductions use
// V_WMMA_F32_16X16X4_F32 (f32 in/out: no precision loss).
// ---------------------------------------------------------------------------

#ifndef __has_builtin
#define __has_builtin(x) 0
#endif

#if __has_builtin(__builtin_amdgcn_global_load_async_to_lds_b128)
#define HAVE_ASYNC 1
#else
#define HAVE_ASYNC 0
#endif

#if __has_builtin(__builtin_amdgcn_wmma_f32_16x16x4_f32)
#define HAVE_WMMA_F32 1
#else
#define HAVE_WMMA_F32 0
#endif

#define CCLS 81      // NUM_CLASSES
#define KMAX 64      // max gt boxes per image
#define TILE 128     // priors per block in the targets pass

typedef float v2f __attribute__((ext_vector_type(2)));
typedef float v8f __attribute__((ext_vector_type(8)));
typedef int   v4i __attribute__((ext_vector_type(4)));

__device__ __forceinline__ float smooth_l1(float d) {
  float a = fabsf(d);
  return (a < 1.0f) ? 0.5f * a * a : a - 0.5f;
}

__device__ __forceinline__ void wait_async0() {
#if HAVE_ASYNC
#if __has_builtin(__builtin_amdgcn_s_wait_asynccnt)
  __builtin_amdgcn_s_wait_asynccnt(0);
#else
  asm volatile("s_wait_asynccnt 0" ::: "memory");
#endif
#endif
}

// Wave(32)-wide f32 sum. Uses V_WMMA_F32_16X16X4_F32 with B = ones:
// A[16x4] holds the 32 lane partials (lane L -> row L%16, K 0 or 2),
// D[m][n] = rowsum(m) for every n; lane l holds 8 rows of column l%16,
// so sum(d[0..7]) + shfl_xor(...,16) = total in every lane. f32 throughout.
__device__ __forceinline__ float wave_reduce_sum(float v) {
#if HAVE_WMMA_F32
  v2f a;    a.x = v;    a.y = 0.0f;
  v2f ones; ones.x = 1.0f; ones.y = 1.0f;
  v8f c = {0.f, 0.f, 0.f, 0.f, 0.f, 0.f, 0.f, 0.f};
  v8f d = __builtin_amdgcn_wmma_f32_16x16x4_f32(
      /*neg_a=*/false, a, /*neg_b=*/false, ones,
      /*c_mod=*/(short)0, c, /*reuse_a=*/false, /*reuse_b=*/false);
  float s = ((d[0] + d[1]) + (d[2] + d[3])) + ((d[4] + d[5]) + (d[6] + d[7]));
  s += __shfl_xor(s, 16, 32);
  return s;
#else
  for (int o = 16; o >= 1; o >>= 1) v += __shfl_xor(v, o, 32);
  return v;
#endif
}

// ---------------------------------------------------------------------------
// K0: zero the u64 best-prior keys and per-batch positive counters.
// ---------------------------------------------------------------------------
__global__ __launch_bounds__(256) void k_init(unsigned long long* keys, int nk,
                                              int* npos, int B) {
  int i = blockIdx.x * blockDim.x + threadIdx.x;
  if (i < nk) keys[i] = 0ull;
  if (i < B) npos[i] = 0;
}

// ---------------------------------------------------------------------------
// K1: matching. One block = 256 priors of one batch image.
//   per-prior: max/argmax IoU over K gts (first-max wins, matching argmax)
//   per-gt   : max over priors via packed (iou_bits<<32 | ~n) u64 max
//              (ties -> smallest n, matching stable argmax), reduced
//              wave-shuffle -> LDS ds_max_u64 -> global atomic max.
// ---------------------------------------------------------------------------
__global__ __launch_bounds__(256) void k_match(
    const float* __restrict__ priors, const float* __restrict__ gt,
    const int* __restrict__ nb, float* __restrict__ ovmax,
    int* __restrict__ ovidx, unsigned long long* __restrict__ keys,
    int N, int K) {
  __shared__ float s_gt[KMAX * 5];
  __shared__ unsigned long long s_key[KMAX];

  const int t = threadIdx.x;
  const int b = blockIdx.y;
  const int n = blockIdx.x * 256 + t;

  for (int i = t; i < K * 5; i += 256) s_gt[i] = gt[(size_t)b * K * 5 + i];
  if (t < K) s_key[t] = 0ull;
  __syncthreads();

  const bool inr = (n < N);
  const int nc = inr ? n : (N - 1);
  const float p0 = priors[nc * 4 + 0], p1 = priors[nc * 4 + 1];
  const float p2 = priors[nc * 4 + 2], p3 = priors[nc * 4 + 3];
  const float pw = p2 - p0 + 1.0f, ph = p3 - p1 + 1.0f;
  const float parea = pw * ph;
  const int nbv = nb[b];

  float best = -2.0f;
  int bk = 0;
  const unsigned nkey = 0xFFFFFFFFu - (unsigned)n;

  for (int k = 0; k < K; ++k) {
    const bool valid = (k < nbv);  // uniform across the wave
    const float g0 = s_gt[k * 5 + 0], g1 = s_gt[k * 5 + 1];
    const float g2 = s_gt[k * 5 + 2], g3 = s_gt[k * 5 + 3];
    const float gw = g2 - g0 + 1.0f, gh = g3 - g1 + 1.0f;
    const float garea = gw * gh;
    const float iw = fminf(p2, g2) - fmaxf(p0, g0) + 1.0f;
    const float ih = fminf(p3, g3) - fmaxf(p1, g1) + 1.0f;
    const float inter = fmaxf(iw, 0.0f) * fmaxf(ih, 0.0f);
    float iou = inter / (parea + garea - inter);
    iou = valid ? iou : -1.0f;              // padded gt slots never match
    if (iou > best) { best = iou; bk = k; } // strict > : first max wins

    // per-gt max over priors (iou >= 0 for valid, so uint order == float order)
    unsigned long long key =
        (valid && inr)
            ? (((unsigned long long)__float_as_uint(iou) << 32) | nkey)
            : 0ull;
    for (int o = 16; o >= 1; o >>= 1) {
      unsigned long long other = __shfl_xor(key, o, 32);
      if (other > key) key = other;
    }
    if ((t & 31) == 0 && key) atomicMax(&s_key[k], key);
  }

  if (inr) {
    ovmax[(size_t)b * N + n] = best;
    ovidx[(size_t)b * N + n] = bk;
  }
  __syncthreads();
  if (t < K && s_key[t]) atomicMax(&keys[(size_t)b * K + t], s_key[t]);
}

// ---------------------------------------------------------------------------
// K2: force the best prior of every valid gt to overlap=2.0 / idx=k.
// Sequential over k per batch (last k wins on collisions), one thread/batch.
// ---------------------------------------------------------------------------
__global__ __launch_bounds__(64) void k_assign(
    const unsigned long long* __restrict__ keys, const int* __restrict__ nb,
    float* __restrict__ ovmax, int* __restrict__ ovidx, int B, int N, int K) {
  int b = blockIdx.x * blockDim.x + threadIdx.x;
  if (b >= B) return;
  int nbv = nb[b];
  for (int k = 0; k < nbv; ++k) {
    unsigned long long key = keys[(size_t)b * K + k];
    unsigned n = 0xFFFFFFFFu - (unsigned)(key & 0xFFFFFFFFull);
    ovmax[(size_t)b * N + n] = 2.0f;
    ovidx[(size_t)b * N + n] = k;
  }
}

// ---------------------------------------------------------------------------
// K3: the bandwidth pass. One block = TILE(128) priors of one batch.
//  - async-DMA the 128x81 f32 conf tile (one contiguous 41472B region) to LDS
//  - overlap: smooth-L1 loc loss per positive prior (threads 0..127)
//  - then wave-cooperative logsumexp over LDS (lanes own classes, shfl reduce)
//  - rank = pos ? 0 : ce stored; per-block deterministic partials for
//    loss_l and sum_pos(ce); integer atomic for num_pos.
// ---------------------------------------------------------------------------
__global__ __launch_bounds__(256) void k_targets(
    const float* __restrict__ loc, const float* __restrict__ conf,
    const float* __restrict__ priors, const float* __restrict__ gt,
    const float* __restrict__ ovmax, const int* __restrict__ ovidx,
    float* __restrict__ rank, int* __restrict__ npos,
    float* __restrict__ posce_part, float* __restrict__ lossl_part,
    int N, int K, int BPB) {
  __shared__ float s_conf[TILE * CCLS];  // 41472 B
  __shared__ int s_ct[TILE];
  __shared__ int s_pos[TILE];
  __shared__ float s_redA[8];
  __shared__ float s_redB[8];

  const int t = threadIdx.x;
  const int b = blockIdx.y;
  const int n0 = blockIdx.x * TILE;
  const int rows = min(TILE, N - n0);
  const float* gsrc = conf + ((size_t)b * N + n0) * CCLS;

  if (rows == TILE) {
    const int CHUNKS = TILE * CCLS / 4;  // 2592 x b128
#if HAVE_ASYNC
    const v4i* g4 = (const v4i*)gsrc;
    v4i* l4 = (v4i*)s_conf;
    for (int i = t; i < CHUNKS; i += 256) {
      __builtin_amdgcn_global_load_async_to_lds_b128(
          (__attribute__((address_space(1))) v4i*)(g4 + i),
          (__attribute__((address_space(3))) v4i*)(l4 + i), 0, 0);
    }
#else
    for (int i = t; i < CHUNKS; i += 256)
      ((float4*)s_conf)[i] = ((const float4*)gsrc)[i];
#endif
  } else {
    for (int i = t; i < rows * CCLS; i += 256) s_conf[i] = gsrc[i];
  }

  // ---- phase A (overlaps the async DMA): matching targets + smooth-L1 ----
  float partial_l = 0.0f;
  if (t < rows) {
    const int n = n0 + t;
    const int k = ovidx[(size_t)b * N + n];
    const float ov = ovmax[(size_t)b * N + n];
    const float* g = gt + ((size_t)b * K + k) * 5;
    const int ct = (ov < 0.5f) ? 0 : (int)g[4];
    const int pos = (ct > 0);
    s_ct[t] = ct;
    s_pos[t] = pos;
    if (pos) {
      const float p0 = priors[n * 4 + 0], p1 = priors[n * 4 + 1];
      const float p2 = priors[n * 4 + 2], p3 = priors[n * 4 + 3];
      const float pw = p2 - p0 + 1.0f, ph = p3 - p1 + 1.0f;
      const float pcx = p0 + 0.5f * pw, pcy = p1 + 0.5f * ph;
      const float gw = g[2] - g[0] + 1.0f, gh = g[3] - g[1] + 1.0f;
      const float gcx = g[0] + 0.5f * gw, gcy = g[1] + 0.5f * gh;
      const float tx = ((gcx - pcx) / pw) * 10.0f;   // / std 0.1
      const float ty = ((gcy - pcy) / ph) * 10.0f;
      const float tw = logf(gw / pw) * 5.0f;         // / std 0.2
      const float th = logf(gh / ph) * 5.0f;
      const float* L = loc + ((size_t)b * N + n) * 4;
      partial_l = smooth_l1(L[0] - tx) + smooth_l1(L[1] - ty) +
                  smooth_l1(L[2] - tw) + smooth_l1(L[3] - th);
    }
  }

  wait_async0();
  __syncthreads();

  // ---- phase B: wave-cooperative logsumexp over the LDS tile ----
  const int wave = t >> 5, lane = t & 31;
  float posce = 0.0f;
  for (int i = 0; i < 16; ++i) {
    const int p = wave * 16 + i;       // uniform across the wave
    if (p < rows) {
      const float* row = s_conf + p * CCLS;
      const float x0 = row[lane];
      const float x1 = row[lane + 32];
      const bool has2 = lane < (CCLS - 64);  // 81 classes -> lanes 0..16
      const float x2 = has2 ? row[lane + 64] : -3.402823466e38f;
      float m = fmaxf(fmaxf(x0, x1), x2);
      for (int o = 16; o >= 1; o >>= 1) m = fmaxf(m, __shfl_xor(m, o, 32));
      float s = expf(x0 - m) + expf(x1 - m) + (has2 ? expf(x2 - m) : 0.0f);
      for (int o = 16; o >= 1; o >>= 1) s += __shfl_xor(s, o, 32);
      const float ce = m + logf(s) - row[s_ct[p]];
      if (lane == 0) {
        const int pos = s_pos[p];
        rank[(size_t)b * N + n0 + p] = pos ? 0.0f : ce;
        if (pos) posce += ce;
      }
    }
  }

  // ---- phase C: deterministic block reductions (WMMA wave reduce) ----
  const float wl = wave_reduce_sum(partial_l);  // full EXEC here
  if (lane == 0) { s_redA[wave] = wl; s_redB[wave] = posce; }
  __syncthreads();
  if (t == 0) {
    float ll = 0.0f, pc = 0.0f;
    int np = 0;
    for (int w = 0; w < 8; ++w) { ll += s_redA[w]; pc += s_redB[w]; }
    for (int p = 0; p < rows; ++p) np += s_pos[p];
    const int gid = b * BPB + blockIdx.x;
    lossl_part[gid] = ll;
    posce_part[gid] = pc;
    atomicAdd(&npos[b], np);  // integer: deterministic
  }
}

// ---------------------------------------------------------------------------
// K4: exact top-k sum per batch via 4-pass radix select on float bits
// (all ranks >= 0 so uint order == float order). Replaces the reference's
// double argsort: the selected-negatives CE sum only needs the k-th largest
// value T, count(>T) and sum(>T); equal-to-T entries each contribute T.
// ---------------------------------------------------------------------------
__global__ __launch_bounds__(256) void k_select(
    const float* __restrict__ rank, const int* __restrict__ npos,
    const float* __restrict__ posce_part, float* __restrict__ loss_c_b,
    int N, int BPB) {
  __shared__ unsigned bins[256];
  __shared__ float s_red[256];
  __shared__ unsigned s_prefix;
  __shared__ int s_kk;
  __shared__ float s_posce;

  const int b = blockIdx.x;
  const int t = threadIdx.x;
  const float* r = rank + (size_t)b * N;

  // sum of CE over positives (deterministic tree over per-block partials)
  float pc = 0.0f;
  for (int i = t; i < BPB; i += 256) pc += posce_part[b * BPB + i];
  s_red[t] = pc;
  __syncthreads();
  for (int s = 128; s > 0; s >>= 1) {
    if (t < s) s_red[t] += s_red[t + s];
    __syncthreads();
  }
  if (t == 0) {
    s_posce = s_red[0];
    int np = npos[b];
    s_kk = min(3 * np, N - 1);  // num_neg
    s_prefix = 0u;
  }
  __syncthreads();

  for (int pass = 0; pass < 4; ++pass) {
    const int shift = 24 - 8 * pass;
    bins[t] = 0u;
    __syncthreads();
    const unsigned pref = s_prefix;
    const unsigned hm = pass ? (0xFFFFFFFFu << (shift + 8)) : 0u;
    for (int i = t; i < N; i += 256) {
      const unsigned u = __float_as_uint(r[i]);
      if (((u ^ pref) & hm) == 0u) atomicAdd(&bins[(u >> shift) & 0xFFu], 1u);
    }
    __syncthreads();
    if (t == 0) {
      int kk = s_kk;
      int d;
      for (d = 255; d >= 0; --d) {
        const unsigned c = bins[d];
        if ((unsigned)kk <= c) break;
        kk -= (int)c;
      }
      if (d < 0) d = 0;
      s_prefix = pref | ((unsigned)d << shift);
      s_kk = kk;
    }
    __syncthreads();
  }

  const unsigned T = s_prefix;  // bit pattern of the k-th largest rank
  float sum = 0.0f;
  for (int i = t; i < N; i += 256) {
    __builtin_prefetch(&r[i + 2048], 0, 0);
    const unsigned u = __float_as_uint(r[i]);
    if (u > T) sum += r[i];
  }
  s_red[t] = sum;
  __syncthreads();
  for (int s = 128; s > 0; s >>= 1) {
    if (t < s) s_red[t] += s_red[t + s];
    __syncthreads();
  }
  if (t == 0)
    loss_c_b[b] = s_posce + s_red[0] + (float)s_kk * __uint_as_float(T);
}

// ---------------------------------------------------------------------------
// K5: final deterministic reduction -> d_out[2] = {loss_l/n, loss_c/n}
// ---------------------------------------------------------------------------
__global__ __launch_bounds__(256) void k_final(
    const float* __restrict__ lossl_part, const float* __restrict__ loss_c_b,
    const int* __restrict__ npos, float* __restrict__ out, int nparts, int B) {
  __shared__ float s_red[256];
  const int t = threadIdx.x;
  float s = 0.0f;
  for (int i = t; i < nparts; i += 256) s += lossl_part[i];
  s_red[t] = s;
  __syncthreads();
  for (int w = 128; w > 0; w >>= 1) {
    if (t < w) s_red[t] += s_red[t + w];
    __syncthreads();
  }
  if (t == 0) {
    float ll = s_red[0], lc = 0.0f;
    int n = 0;
    for (int b = 0; b < B; ++b) { lc += loss_c_b[b]; n += npos[b]; }
    const float inv = 1.0f / (float)n;  // n >= 1 (num_boxes >= 1 -> forced pos)
    out[0] = ll * inv;
    out[1] = lc * inv;
  }
}

// ---------------------------------------------------------------------------
extern "C" void kernel_launch(void* const* d_in, const int* in_sizes, int n_in,
                              void* d_out, int out_size, void* d_ws,
                              size_t ws_size, hipStream_t stream) {
  const float* loc = (const float*)d_in[0];
  const float* conf = (const float*)d_in[1];
  const float* priors = (const float*)d_in[2];
  const float* gt = (const float*)d_in[3];
  const int* nb = (const int*)d_in[4];

  const int B = in_sizes[4];
  const int N = in_sizes[2] / 4;
  const int K = in_sizes[3] / (B * 5);
  const int BPB = (N + TILE - 1) / TILE;  // K3/K4 blocks per batch

  // ---- workspace carve (≈12 MB) ----
  size_t off = 0;
  auto take = [&](size_t bytes) -> void* {
    off = (off + 255) & ~(size_t)255;
    void* p = (char*)d_ws + off;
    off += bytes;
    return p;
  };
  unsigned long long* keys = (unsigned long long*)take((size_t)B * K * 8);
  float* ovmax = (float*)take((size_t)B * N * 4);
  int* ovidx = (int*)take((size_t)B * N * 4);
  float* rankb = (float*)take((size_t)B * N * 4);
  int* npos = (int*)take((size_t)B * 4);
  float* posce_part = (float*)take((size_t)B * BPB * 4);
  float* lossl_part = (float*)take((size_t)B * BPB * 4);
  float* loss_c_b = (float*)take((size_t)B * 4);
  (void)ws_size; (void)n_in; (void)out_size;

  k_init<<<(B * K + 255) / 256, 256, 0, stream>>>(keys, B * K, npos, B);
  k_match<<<dim3((N + 255) / 256, B), 256, 0, stream>>>(priors, gt, nb, ovmax,
                                                        ovidx, keys, N, K);
  k_assign<<<(B + 63) / 64, 64, 0, stream>>>(keys, nb, ovmax, ovidx, B, N, K);
  k_targets<<<dim3(BPB, B), 256, 0, stream>>>(loc, conf, priors, gt, ovmax,
                                              ovidx, rankb, npos, posce_part,
                                              lossl_part, N, K, BPB);
  k_select<<<B, 256, 0, stream>>>(rankb, npos, posce_part, loss_c_b, N, BPB);
  k_final<<<1, 256, 0, stream>>>(lossl_part, loss_c_b, npos, (float*)d_out,
                                 B * BPB, B);
}